// GATLayer_83992380440763
// MI455X (gfx1250) — compile-verified
//
#include <hip/hip_runtime.h>
#include <hip/hip_bf16.h>

#define N_NODES   50000
#define E_EDGES   1000000
#define IN_DIM    128
#define OUT_DIM   64
#define NEG_SLOPE 0.01f

typedef __attribute__((ext_vector_type(2))) float v2f;
typedef __attribute__((ext_vector_type(8))) float v8f;

// Monotonic uint encoding of float for atomicMax-based segment max.
__device__ __forceinline__ unsigned f32_ord(float f) {
  unsigned u = __float_as_uint(f);
  return (u & 0x80000000u) ? ~u : (u | 0x80000000u);
}
__device__ __forceinline__ float ord_f32(unsigned o) {
  unsigned u = (o & 0x80000000u) ? (o & 0x7FFFFFFFu) : ~o;
  return __uint_as_float(u);
}

// ---------------------------------------------------------------------------
// K1: z[N,64] = h[N,128] @ Wfc[64,128]^T via V_WMMA_F32_16X16X4_F32.
// One wave per 16-row M-tile; 4 accumulators cover all 64 output columns.
// A (16x4 f32): lane<16 -> K=k..k+1, lane>=16 -> K=k+2..k+3, M = lane&15.
// B (4x16 f32): same K split, N = lane&15 (B[k][n] = Wfc[n][k]).
// C/D (16x16 f32): vgpr r, lane<16 -> M=r, lane>=16 -> M=r+8, N = lane&15.
// ---------------------------------------------------------------------------
__global__ void __launch_bounds__(256)
gat_gemm_wmma(const float* __restrict__ h, const float* __restrict__ Wfc,
              float* __restrict__ z) {
  const int lane = threadIdx.x & 31;
  const int wid  = blockIdx.x * (blockDim.x >> 5) + (threadIdx.x >> 5);
  const int m0   = wid * 16;
  if (m0 >= N_NODES) return;

  const int lrow = lane & 15;           // M (for A/C) or N (for B) sub-index
  const int kh   = (lane >> 4) << 1;    // K sub-offset: 0 or 2

  const float* __restrict__ ap  = h   + (long)(m0 + lrow) * IN_DIM + kh;
  const float* __restrict__ b0p = Wfc + (long)( 0 + lrow) * IN_DIM + kh;
  const float* __restrict__ b1p = Wfc + (long)(16 + lrow) * IN_DIM + kh;
  const float* __restrict__ b2p = Wfc + (long)(32 + lrow) * IN_DIM + kh;
  const float* __restrict__ b3p = Wfc + (long)(48 + lrow) * IN_DIM + kh;

  v8f c0 = {}, c1 = {}, c2 = {}, c3 = {};
  #pragma unroll 4
  for (int k = 0; k < IN_DIM; k += 4) {
    v2f a  = *(const v2f*)(ap  + k);
    v2f b0 = *(const v2f*)(b0p + k);
    v2f b1 = *(const v2f*)(b1p + k);
    v2f b2 = *(const v2f*)(b2p + k);
    v2f b3 = *(const v2f*)(b3p + k);
    c0 = __builtin_amdgcn_wmma_f32_16x16x4_f32(false, a, false, b0, (short)0, c0, false, false);
    c1 = __builtin_amdgcn_wmma_f32_16x16x4_f32(false, a, false, b1, (short)0, c1, false, false);
    c2 = __builtin_amdgcn_wmma_f32_16x16x4_f32(false, a, false, b2, (short)0, c2, false, false);
    c3 = __builtin_amdgcn_wmma_f32_16x16x4_f32(false, a, false, b3, (short)0, c3, false, false);
  }

  const int rbase = m0 + ((lane >> 4) << 3);     // M = r or r+8 per half-wave
  float* zp = z + (long)rbase * OUT_DIM + lrow;
  #pragma unroll
  for (int r = 0; r < 8; ++r) {
    zp[(long)r * OUT_DIM +  0] = c0[r];
    zp[(long)r * OUT_DIM + 16] = c1[r];
    zp[(long)r * OUT_DIM + 32] = c2[r];
    zp[(long)r * OUT_DIM + 48] = c3[r];
  }
}

// ---------------------------------------------------------------------------
// K2: per-node attention scalars zs[n] = z[n].a_src, zd[n] = z[n].a_dst.
// One wave per node; 2 elements per lane; wave32 xor-shuffle reduction.
// ---------------------------------------------------------------------------
__global__ void __launch_bounds__(256)
gat_node_dots(const float* __restrict__ z, const float* __restrict__ Wattn,
              float* __restrict__ zs, float* __restrict__ zd) {
  const int lane = threadIdx.x & 31;
  const int n = blockIdx.x * (blockDim.x >> 5) + (threadIdx.x >> 5);
  if (n >= N_NODES) return;
  const float* zr = z + (long)n * OUT_DIM;
  float z0 = zr[lane], z1 = zr[lane + 32];
  float ps = z0 * Wattn[lane]           + z1 * Wattn[lane + 32];
  float pd = z0 * Wattn[OUT_DIM + lane] + z1 * Wattn[OUT_DIM + lane + 32];
  #pragma unroll
  for (int o = 16; o > 0; o >>= 1) {
    ps += __shfl_xor(ps, o, 32);
    pd += __shfl_xor(pd, o, 32);
  }
  if (lane == 0) { zs[n] = ps; zd[n] = pd; }
}

// K3: zero out[], init segment max (-inf encoded) and segment sum.
__global__ void gat_init(float* __restrict__ out, unsigned* __restrict__ mord,
                         float* __restrict__ ssum) {
  long i = (long)blockIdx.x * blockDim.x + threadIdx.x;
  if (i < (long)N_NODES * OUT_DIM) out[i] = 0.0f;
  if (i < N_NODES) { mord[i] = 0x007FFFFFu /* ord(-inf) */; ssum[i] = 0.0f; }
}

// K4: per-edge logit e = w * leaky_relu(zs[src]+zd[dst]); segment max via
// atomicMax on monotonic-uint encoding.
__global__ void gat_edge_scores(const int* __restrict__ src, const int* __restrict__ dst,
                                const float* __restrict__ wgt,
                                const float* __restrict__ zs, const float* __restrict__ zd,
                                float* __restrict__ ew, unsigned* __restrict__ mord) {
  int i = blockIdx.x * blockDim.x + threadIdx.x;
  if (i >= E_EDGES) return;
  int s0 = src[i], d0 = dst[i];
  float x = zs[s0] + zd[d0];
  float e = (x > 0.0f) ? x : NEG_SLOPE * x;
  e *= wgt[i];
  ew[i] = e;
  atomicMax(mord + d0, f32_ord(e));
}

// K5: ex = exp(e - m[dst]); overwrite ew with ex; segment sum via atomicAdd.
__global__ void gat_edge_exp(const int* __restrict__ dst, const unsigned* __restrict__ mord,
                             float* __restrict__ ew, float* __restrict__ ssum) {
  int i = blockIdx.x * blockDim.x + threadIdx.x;
  if (i >= E_EDGES) return;
  int d0 = dst[i];
  float ex = expf(ew[i] - ord_f32(mord[d0]));
  ew[i] = ex;
  atomicAdd(ssum + d0, ex);
}

// K6: out[dst] += (ex/s[dst]) * z[src]. One wave per edge; float2 per lane;
// z and out are L2-resident (12.8 MB each vs 192 MB L2).
__global__ void __launch_bounds__(256)
gat_aggregate(const int* __restrict__ src, const int* __restrict__ dst,
              const float* __restrict__ ew, const float* __restrict__ ssum,
              const float* __restrict__ z, float* __restrict__ out) {
  const int lane = threadIdx.x & 31;
  const int e = blockIdx.x * (blockDim.x >> 5) + (threadIdx.x >> 5);
  if (e >= E_EDGES) return;
  const int s0 = src[e], d0 = dst[e];
  const float alpha = ew[e] / ssum[d0];
  v2f v = *(const v2f*)(z + (long)s0 * OUT_DIM + lane * 2);
  float* op = out + (long)d0 * OUT_DIM + lane * 2;
  atomicAdd(op + 0, alpha * v.x);
  atomicAdd(op + 1, alpha * v.y);
}

extern "C" void kernel_launch(void* const* d_in, const int* in_sizes, int n_in,
                              void* d_out, int out_size, void* d_ws, size_t ws_size,
                              hipStream_t stream) {
  (void)in_sizes; (void)n_in; (void)out_size; (void)ws_size;
  const float* h     = (const float*)d_in[0];
  const float* Wfc   = (const float*)d_in[1];
  const float* Wattn = (const float*)d_in[2];
  const int*   src   = (const int*)d_in[3];
  const int*   dst   = (const int*)d_in[4];
  const float* wgt   = (const float*)d_in[5];
  float* out = (float*)d_out;

  // Workspace layout (~17.6 MB total)
  float*    z    = (float*)d_ws;                         // N*64
  float*    zs   = z + (size_t)N_NODES * OUT_DIM;        // N
  float*    zd   = zs + N_NODES;                         // N
  unsigned* mord = (unsigned*)(zd + N_NODES);            // N
  float*    ssum = (float*)(mord + N_NODES);             // N
  float*    ew   = ssum + N_NODES;                       // E

  {
    int waves  = (N_NODES + 15) / 16;                    // 3125
    int blocks = (waves * 32 + 255) / 256;
    gat_gemm_wmma<<<blocks, 256, 0, stream>>>(h, Wfc, z);
  }
  {
    int blocks = (N_NODES * 32 + 255) / 256;
    gat_node_dots<<<blocks, 256, 0, stream>>>(z, Wattn, zs, zd);
  }
  {
    long total = (long)N_NODES * OUT_DIM;
    int blocks = (int)((total + 255) / 256);
    gat_init<<<blocks, 256, 0, stream>>>(out, mord, ssum);
  }
  {
    int blocks = (E_EDGES + 255) / 256;
    gat_edge_scores<<<blocks, 256, 0, stream>>>(src, dst, wgt, zs, zd, ew, mord);
    gat_edge_exp<<<blocks, 256, 0, stream>>>(dst, mord, ew, ssum);
  }
  {
    long threads = (long)E_EDGES * 32;
    int blocks = (int)((threads + 255) / 256);
    gat_aggregate<<<blocks, 256, 0, stream>>>(src, dst, ew, ssum, z, out);
  }
}